// GAT_43499428774061
// MI455X (gfx1250) — compile-verified
//
#include <hip/hip_runtime.h>
#include <math.h>

#define FEAT 256
#define NEG_SLOPE 0.2f

typedef __attribute__((ext_vector_type(2))) float v2f;
typedef __attribute__((ext_vector_type(8))) float v8f;

// ---------------------------------------------------------------------------
// GEMM: C[M,Ncols] = op(A)[M,K] @ B[K,Ncols], op = optional ReLU (compile-time).
// One wave32 per 16x16 output tile using V_WMMA_F32_16X16X4_F32 (full fp32).
// fp32 A-matrix 16x4 layout: lanes 0-15 hold K={0,1}, lanes 16-31 hold K={2,3},
// row M = lane&15.  B mirrored with N in lanes.  C/D: v8f, row = r + 8*(lane>>4).
// ---------------------------------------------------------------------------
template<bool RELU>
__global__ __launch_bounds__(32)
void gemm_wmma_f32(const float* __restrict__ A, const float* __restrict__ B,
                   float* __restrict__ C, int M, int K, int Ncols)
{
  const int lane = threadIdx.x;
  const int half = lane >> 4;   // 0: K pair {0,1}, 1: K pair {2,3}
  const int l    = lane & 15;
  const int tileM = blockIdx.x * 16;
  const int tileN = blockIdx.y * 16;

  int arow = tileM + l; if (arow >= M) arow = M - 1;   // clamp, no EXEC divergence
  const float* Ap = A + (long)arow * K + 2 * half;
  const float* Bp = B + (long)(2 * half) * Ncols + tileN + l;

  v8f acc = {0.f, 0.f, 0.f, 0.f, 0.f, 0.f, 0.f, 0.f};
  #pragma unroll 4
  for (int k = 0; k < K; k += 4) {
    float2 a2 = *(const float2*)(Ap + k);              // contiguous K pair
    if (RELU) { a2.x = fmaxf(a2.x, 0.f); a2.y = fmaxf(a2.y, 0.f); }
    float b0 = Bp[(long)k * Ncols];
    float b1 = Bp[(long)(k + 1) * Ncols];
    v2f av = {a2.x, a2.y};
    v2f bv = {b0, b1};
    acc = __builtin_amdgcn_wmma_f32_16x16x4_f32(false, av, false, bv,
                                                (short)0, acc, false, false);
  }
  #pragma unroll
  for (int r = 0; r < 8; ++r) {
    int row = tileM + r + 8 * half;
    if (row < M) C[(long)row * Ncols + tileN + l] = acc[r];
  }
}

// ---------------------------------------------------------------------------
// Per-(node,head) attention dots: a_src[n,h] = <h[n,h,:], att_src[h,:]>, same dst.
// ---------------------------------------------------------------------------
__global__ void att_dot(const float* __restrict__ hmat,
                        const float* __restrict__ att_s,
                        const float* __restrict__ att_d,
                        float* __restrict__ a_s, float* __restrict__ a_d,
                        int Nn, int H, int C)
{
  int t = blockIdx.x * blockDim.x + threadIdx.x;
  if (t >= Nn * H) return;
  int n = t / H, h = t % H;
  const float* hp = hmat + (long)n * H * C + (long)h * C;
  const float* sp = att_s + (long)h * C;
  const float* dp = att_d + (long)h * C;
  float s = 0.f, d = 0.f;
  for (int c = 0; c < C; c += 4) {
    float4 hv = *(const float4*)(hp + c);
    float4 sv = *(const float4*)(sp + c);
    float4 dv = *(const float4*)(dp + c);
    s += hv.x * sv.x + hv.y * sv.y + hv.z * sv.z + hv.w * sv.w;
    d += hv.x * dv.x + hv.y * dv.y + hv.z * dv.z + hv.w * dv.w;
  }
  a_s[t] = s; a_d[t] = d;
}

__global__ void fill_f32(float* __restrict__ p, float v, int n)
{
  int t = blockIdx.x * blockDim.x + threadIdx.x;
  if (t < n) p[t] = v;
}

// out[n, j] = bias[j]   (broadcast init before edge scatter)
__global__ void bias_init(float* __restrict__ out, const float* __restrict__ bias,
                          int Nn, int HC)
{
  int t = blockIdx.x * blockDim.x + threadIdx.x;
  if (t < Nn * HC) out[t] = bias[t % HC];
}

__device__ __forceinline__ void atomicMaxF(float* addr, float val)
{
  // exact float-max via sign-split integer atomics (init must be -inf)
  if (!(val < 0.f)) atomicMax((int*)addr, __float_as_int(val));
  else              atomicMin((unsigned int*)addr, __float_as_uint(val));
}

__device__ __forceinline__ void edge_endpoints(const int* __restrict__ ei,
                                               int E, int e, int& src, int& dst)
{
  if (e < E) { src = ei[e]; dst = ei[E + e]; } else { src = dst = e - E; }
}

// alpha_raw = leaky_relu(a_src[src]+a_dst[dst]); segment max into amax[dst].
// One thread per edge; heads unrolled at compile time.
template<int H>
__global__ void edge_attn_max(const float* __restrict__ a_s,
                              const float* __restrict__ a_d,
                              const int* __restrict__ ei,
                              int E, int Etot,
                              float* __restrict__ alpha, float* __restrict__ amax)
{
  int e = blockIdx.x * blockDim.x + threadIdx.x;
  if (e >= Etot) return;
  int src, dst; edge_endpoints(ei, E, e, src, dst);
  #pragma unroll
  for (int h = 0; h < H; ++h) {
    float v = a_s[src * H + h] + a_d[dst * H + h];
    v = v > 0.f ? v : v * NEG_SLOPE;
    alpha[(long)e * H + h] = v;
    atomicMaxF(amax + dst * H + h, v);
  }
}

// alpha = exp(alpha_raw - amax[dst]); segment sum into denom[dst]
template<int H>
__global__ void edge_expsum(const int* __restrict__ ei, int E, int Etot,
                            const float* __restrict__ amax,
                            float* __restrict__ alpha, float* __restrict__ denom)
{
  int e = blockIdx.x * blockDim.x + threadIdx.x;
  if (e >= Etot) return;
  int src, dst; edge_endpoints(ei, E, e, src, dst);
  #pragma unroll
  for (int h = 0; h < H; ++h) {
    float ex = expf(alpha[(long)e * H + h] - amax[dst * H + h]);
    alpha[(long)e * H + h] = ex;
    atomicAdd(denom + dst * H + h, ex);
  }
}

// alpha /= (denom[dst] + eps)  — hoists the divide out of the wide scatter
template<int H>
__global__ void edge_norm(const int* __restrict__ ei, int E, int Etot,
                          const float* __restrict__ denom,
                          float* __restrict__ alpha)
{
  int e = blockIdx.x * blockDim.x + threadIdx.x;
  if (e >= Etot) return;
  int src, dst; edge_endpoints(ei, E, e, src, dst);
  #pragma unroll
  for (int h = 0; h < H; ++h)
    alpha[(long)e * H + h] /= (denom[dst * H + h] + 1e-16f);
}

// out[dst] += h[src] * alpha.  threadIdx.x = 4-channel chunk,
// threadIdx.y = edge-within-block (block = 256 threads).
__global__ void scatter_msg(const float* __restrict__ hmat,
                            const float* __restrict__ alpha,
                            const int* __restrict__ ei,
                            int E, int Etot, int H, int C,
                            float* __restrict__ out)
{
  int e = blockIdx.x * blockDim.y + threadIdx.y;
  if (e >= Etot) return;
  int src, dst; edge_endpoints(ei, E, e, src, dst);
  int hc   = threadIdx.x * 4;
  int head = hc / C;
  float w = alpha[(long)e * H + head];
  float4 hv = *(const float4*)(hmat + (long)src * H * C + hc);
  float* o = out + (long)dst * H * C + hc;
  atomicAdd(o + 0, hv.x * w);
  atomicAdd(o + 1, hv.y * w);
  atomicAdd(o + 2, hv.z * w);
  atomicAdd(o + 3, hv.w * w);
}

// row-wise log_softmax over 64 channels, one thread per node (row in registers)
__global__ void logsoftmax64(const float* __restrict__ in, float* __restrict__ out,
                             int Nn)
{
  int n = blockIdx.x * blockDim.x + threadIdx.x;
  if (n >= Nn) return;
  const float* p = in + (long)n * 64;
  float vals[64];
  float m = -INFINITY;
  #pragma unroll
  for (int c = 0; c < 64; c += 4) {
    float4 v = *(const float4*)(p + c);
    vals[c] = v.x; vals[c + 1] = v.y; vals[c + 2] = v.z; vals[c + 3] = v.w;
    m = fmaxf(m, fmaxf(fmaxf(v.x, v.y), fmaxf(v.z, v.w)));
  }
  float s = 0.f;
  #pragma unroll
  for (int c = 0; c < 64; ++c) s += expf(vals[c] - m);
  float lse = m + logf(s);
  float* o = out + (long)n * 64;
  #pragma unroll
  for (int c = 0; c < 64; ++c) o[c] = vals[c] - lse;
}

// ---------------------------------------------------------------------------
extern "C" void kernel_launch(void* const* d_in, const int* in_sizes, int n_in,
                              void* d_out, int out_size, void* d_ws, size_t ws_size,
                              hipStream_t stream)
{
  const float* x     = (const float*)d_in[0];
  const int*   ei    = (const int*)  d_in[1];   // [2,E] int32: row0=src, row1=dst
  const float* W1m   = (const float*)d_in[2];   // [256,128]
  const float* atts1 = (const float*)d_in[3];   // [2,64]
  const float* attd1 = (const float*)d_in[4];
  const float* b1    = (const float*)d_in[5];   // [128]
  const float* W2m   = (const float*)d_in[6];   // [128,64]
  const float* atts2 = (const float*)d_in[7];   // [1,64]
  const float* attd2 = (const float*)d_in[8];
  const float* b2    = (const float*)d_in[9];   // [64]

  const int Nn   = in_sizes[0] / FEAT;   // 50000
  const int E    = in_sizes[1] / 2;      // 1.6M
  const int Etot = E + Nn;               // + self loops

  // ---- workspace carve-out ----
  char* ws = (char*)d_ws;
  size_t off = 0;
  auto alloc = [&](size_t bytes) -> void* {
    void* p = ws + off;
    off += (bytes + 255) & ~(size_t)255;
    return p;
  };
  float* h1     = (float*)alloc((size_t)Nn * 128 * 4);   // x@W1     [N,2,64]
  float* as1    = (float*)alloc((size_t)Nn * 2 * 4);
  float* ad1    = (float*)alloc((size_t)Nn * 2 * 4);
  float* amax1  = (float*)alloc((size_t)Nn * 2 * 4);
  float* den1   = (float*)alloc((size_t)Nn * 2 * 4);
  float* alpha1 = (float*)alloc((size_t)Etot * 2 * 4);
  float* o1     = (float*)alloc((size_t)Nn * 128 * 4);   // layer-1 output
  float* h2     = (float*)alloc((size_t)Nn * 64 * 4);    // relu(o1)@W2
  float* as2    = (float*)alloc((size_t)Nn * 4);
  float* ad2    = (float*)alloc((size_t)Nn * 4);
  float* amax2  = (float*)alloc((size_t)Nn * 4);
  float* den2   = (float*)alloc((size_t)Nn * 4);
  float* alpha2 = (float*)alloc((size_t)Etot * 4);
  float* o2     = (float*)alloc((size_t)Nn * 64 * 4);    // layer-2 output

  const int TB = 256;
  auto cdiv = [](long a, long b) { return (int)((a + b - 1) / b); };

  // ================= Layer 1 (H=2, C=64) =================
  gemm_wmma_f32<false><<<dim3(cdiv(Nn, 16), 8), 32, 0, stream>>>(x, W1m, h1, Nn, 256, 128);
  att_dot<<<cdiv((long)Nn * 2, TB), TB, 0, stream>>>(h1, atts1, attd1, as1, ad1, Nn, 2, 64);
  fill_f32<<<cdiv((long)Nn * 2, TB), TB, 0, stream>>>(amax1, -INFINITY, Nn * 2);
  fill_f32<<<cdiv((long)Nn * 2, TB), TB, 0, stream>>>(den1, 0.f, Nn * 2);
  edge_attn_max<2><<<cdiv(Etot, TB), TB, 0, stream>>>(as1, ad1, ei, E, Etot, alpha1, amax1);
  edge_expsum<2><<<cdiv(Etot, TB), TB, 0, stream>>>(ei, E, Etot, amax1, alpha1, den1);
  edge_norm<2><<<cdiv(Etot, TB), TB, 0, stream>>>(ei, E, Etot, den1, alpha1);
  bias_init<<<cdiv((long)Nn * 128, TB), TB, 0, stream>>>(o1, b1, Nn, 128);
  // H*C = 128 -> 32 chunks/edge, 8 edges/block
  scatter_msg<<<cdiv(Etot, 8), dim3(32, 8), 0, stream>>>(h1, alpha1, ei,
                                                         E, Etot, 2, 64, o1);

  // ================= Layer 2 (H=1, C=64), ReLU fused into GEMM A-load =========
  gemm_wmma_f32<true><<<dim3(cdiv(Nn, 16), 4), 32, 0, stream>>>(o1, W2m, h2, Nn, 128, 64);
  att_dot<<<cdiv((long)Nn, TB), TB, 0, stream>>>(h2, atts2, attd2, as2, ad2, Nn, 1, 64);
  fill_f32<<<cdiv((long)Nn, TB), TB, 0, stream>>>(amax2, -INFINITY, Nn);
  fill_f32<<<cdiv((long)Nn, TB), TB, 0, stream>>>(den2, 0.f, Nn);
  edge_attn_max<1><<<cdiv(Etot, TB), TB, 0, stream>>>(as2, ad2, ei, E, Etot, alpha2, amax2);
  edge_expsum<1><<<cdiv(Etot, TB), TB, 0, stream>>>(ei, E, Etot, amax2, alpha2, den2);
  edge_norm<1><<<cdiv(Etot, TB), TB, 0, stream>>>(ei, E, Etot, den2, alpha2);
  bias_init<<<cdiv((long)Nn * 64, TB), TB, 0, stream>>>(o2, b2, Nn, 64);
  // H*C = 64 -> 16 chunks/edge, 16 edges/block
  scatter_msg<<<cdiv(Etot, 16), dim3(16, 16), 0, stream>>>(h2, alpha2, ei,
                                                           E, Etot, 1, 64, o2);

  // ================= log_softmax -> d_out =================
  logsoftmax64<<<cdiv((long)Nn, TB), TB, 0, stream>>>(o2, (float*)d_out, Nn);
}